// XYHamiltonian_88691074662571
// MI455X (gfx1250) — compile-verified
//
#include <hip/hip_runtime.h>

// XY-model log-density, MI455X (gfx1250, wave32).
// One 256-thread block per sample row (4096 angles, 16 KB staged in LDS).
// Bulk math: native v_cos_f32 (memory-bound-balanced). Wave reduction: two
// V_WMMA_F32_16X16X4_F32 with an all-ones B matrix (exact f32, shuffle-free).

#define LENGTH   64
#define LATTICE  4096
#define NTHREADS 256
#define NWAVES   (NTHREADS / 32)
#define SITES_PER_THREAD (LATTICE / NTHREADS)   // 16
#define VEC4_PER_THREAD  (LATTICE / 4 / NTHREADS) // 4

typedef float v2f __attribute__((ext_vector_type(2)));
typedef float v8f __attribute__((ext_vector_type(8)));

__global__ __launch_bounds__(NTHREADS) void xy_logdensity_kernel(
    const float* __restrict__ state, float* __restrict__ out)
{
    __shared__ float th[LATTICE];
    __shared__ float wsum[NWAVES];

    const int tid = threadIdx.x;

    // ---- Stage one sample row (16 KB) into LDS, coalesced float4 ----
    const float4* row4 = reinterpret_cast<const float4*>(
        state + (size_t)blockIdx.x * LATTICE);
    float4* th4 = reinterpret_cast<float4*>(th);
#pragma unroll
    for (int k = 0; k < VEC4_PER_THREAD; ++k)
        th4[tid + NTHREADS * k] = row4[tid + NTHREADS * k];
    __syncthreads();

    // ---- Per-thread accumulation over 16 sites, 2 bonds each ----
    // site i = r*64 + c ; up = ((r+1)%64, c) = (i+64) mod 4096
    //                     right = (r, (c+1)%64) = (i & ~63) | ((i+1) & 63)
    float acc = 0.0f;
#pragma unroll
    for (int k = 0; k < SITES_PER_THREAD; ++k) {
        const int i  = tid + NTHREADS * k;
        const float ti = th[i];
        const float tu = th[(i + LENGTH) & (LATTICE - 1)];
        const float tr = th[(i & ~(LENGTH - 1)) | ((i + 1) & (LENGTH - 1))];
        acc += __cosf(tu - ti) + __cosf(tr - ti);   // v_cos_f32 (native)
    }

    // ---- 32-lane reduction via V_WMMA_F32_16X16X4_F32, B = ones ----
    // A layout (16x4 f32): lane L<16: vgpr0=A[L,0], vgpr1=A[L,1];
    //                      lane L>=16: vgpr0=A[L-16,2], vgpr1=A[L-16,3].
    // With a = {acc, 0}: D[m,n] = p[m] + p[m+16] = q[m] for every n.
    // D layout: lane L<16 holds q[0..7] in its 8 vgprs, L>=16 holds q[8..15].
    // Per-lane 8-way add + second ones-WMMA -> full wave sum in every lane.
    float wave_total;
#if __has_builtin(__builtin_amdgcn_wmma_f32_16x16x4_f32)
    {
        v2f a; a[0] = acc;  a[1] = 0.0f;
        v2f b; b[0] = 1.0f; b[1] = 1.0f;
        v8f c = {};
        v8f d = __builtin_amdgcn_wmma_f32_16x16x4_f32(
            false, a, false, b, (short)0, c, false, false);
        float r = d[0] + d[1] + d[2] + d[3] + d[4] + d[5] + d[6] + d[7];
        v2f a2; a2[0] = r; a2[1] = 0.0f;
        v8f d2 = __builtin_amdgcn_wmma_f32_16x16x4_f32(
            false, a2, false, b, (short)0, c, false, false);
        wave_total = d2[0];
    }
#else
    wave_total = acc;
    for (int off = 16; off > 0; off >>= 1)
        wave_total += __shfl_xor(wave_total, off, 32);
#endif

    // ---- Cross-wave combine (8 values) ----
    const int lane = tid & 31;
    const int wave = tid >> 5;
    if (lane == 0) wsum[wave] = wave_total;
    __syncthreads();
    if (tid == 0) {
        float t = 0.0f;
#pragma unroll
        for (int w = 0; w < NWAVES; ++w) t += wsum[w];
        out[blockIdx.x] = t;   // BETA == 1.0
    }
}

extern "C" void kernel_launch(void* const* d_in, const int* in_sizes, int n_in,
                              void* d_out, int out_size, void* d_ws, size_t ws_size,
                              hipStream_t stream)
{
    (void)in_sizes; (void)n_in; (void)d_ws; (void)ws_size;
    const float* state = (const float*)d_in[0];
    // d_in[1] (shift, int64) intentionally unused: the periodic-lattice
    // neighbour indices are computed analytically in-kernel and match
    // _make_shift() exactly (roll -1 along rows and columns of a 64x64 grid).
    float* out = (float*)d_out;
    const int nsamples = out_size;   // 16384 rows, one block each
    xy_logdensity_kernel<<<nsamples, NTHREADS, 0, stream>>>(state, out);
}